// GraphAggregateLayers_32993938768351
// MI455X (gfx1250) — compile-verified
//
#include <hip/hip_runtime.h>
#include <hip/hip_bf16.h>
#include <float.h>

typedef __attribute__((ext_vector_type(2))) float v2f;
typedef __attribute__((ext_vector_type(8))) float v8f;

#define D_EMB   64
#define HEADS   4
#define WAVES_PER_BLOCK 4
#define EDGES_PER_WAVE  16
#define EDGES_PER_BLOCK (WAVES_PER_BLOCK * EDGES_PER_WAVE)
// W stored row-pair interleaved: element (r,c) -> [(r>>1)*SW2 + 2*c + (r&1)].
// SW2 = 160 floats (mod 64 = 32) -> b64 B-frag loads are bank-conflict-free
// (low half banks 2c..2c+1, high half offset by 32 banks).
#define SW2 160
#define SA_STRIDE 68   // A tile LDS row stride (floats): conflict-free b64 A-frag reads

// ---- float atomic max via monotone int/uint punning (works with -FLT_MAX init) ----
__device__ __forceinline__ void atomicMaxF(float* addr, float v) {
    if (v >= 0.0f) atomicMax((int*)addr, __float_as_int(v));
    else           atomicMin((unsigned int*)addr, __float_as_uint(v));
}

// ============================================================================
// Kernel 1: per-edge-tile fp32 WMMA GEMMs + per-head dot -> raw_scores [E,4]
//           and per-(item,head) segment max via atomics.
// Each wave handles 16 edges: Q = (rel*ent) @ W_Q, K = ent[item] @ W_K,
// raw[e,h] = dot(Q[e,h*16:(h+1)*16], K[e,...]) / sqrt(16).
// ============================================================================
__global__ __launch_bounds__(128) void kgemm_scores(
    const int* __restrict__ item_ids, const int* __restrict__ rel_ids,
    const int* __restrict__ ent_ids,
    const float* __restrict__ ent_embs, const float* __restrict__ rel_embs,
    const float* __restrict__ WQ, const float* __restrict__ WK,
    float* __restrict__ raw_out, float* __restrict__ seg_max, long E)
{
    __shared__ float sWQ[(D_EMB / 2) * SW2];
    __shared__ float sWK[(D_EMB / 2) * SW2];
    __shared__ float sAq[WAVES_PER_BLOCK][EDGES_PER_WAVE * SA_STRIDE];
    __shared__ float sAk[WAVES_PER_BLOCK][EDGES_PER_WAVE * SA_STRIDE];

    const int tid  = threadIdx.x;
    const int lane = tid & 31;
    const int wave = tid >> 5;

    // ---- stage W_Q / W_K (64x64) into pair-interleaved LDS, cooperatively ----
    for (int i = tid; i < D_EMB * D_EMB; i += 128) {
        int r = i >> 6, c = i & 63;
        int d = (r >> 1) * SW2 + (c << 1) + (r & 1);
        sWQ[d] = WQ[i];
        sWK[d] = WK[i];
    }

    // ---- stage A tiles: 16 rows of 64; lane half loads 32 cols of one row ----
    const long base = (long)blockIdx.x * EDGES_PER_BLOCK + (long)wave * EDGES_PER_WAVE;
    const int row  = lane & 15;
    const int half = lane >> 4;
    {
        long e = base + row;
        if (e >= E) e = E - 1;                       // clamp loads; stores guarded later
        const int re = rel_ids[e];
        const int en = ent_ids[e];
        const int it = item_ids[e];
        const float4* relp = (const float4*)(rel_embs + (long)re * D_EMB);
        const float4* entp = (const float4*)(ent_embs + (long)en * D_EMB);
        const float4* itp  = (const float4*)(ent_embs + (long)it * D_EMB);
        float* aq = &sAq[wave][row * SA_STRIDE];
        float* ak = &sAk[wave][row * SA_STRIDE];
#pragma unroll
        for (int j = 0; j < 8; ++j) {
            int c4 = half * 8 + j;                   // float4 index 0..15
            float4 r4 = relp[c4];
            float4 e4 = entp[c4];
            float4 q4;
            q4.x = r4.x * e4.x; q4.y = r4.y * e4.y;
            q4.z = r4.z * e4.z; q4.w = r4.w * e4.w;
            *(float4*)&aq[c4 * 4] = q4;
            *(float4*)&ak[c4 * 4] = itp[c4];
        }
    }
    __syncthreads();

    // ---- WMMA: D(16x64) = A(16x64) x W(64x64), fp32, K stepped by 4 ----
    // A frag (16x4 f32): lane<16 -> row=lane, K = 4s+0,4s+1 ; lane>=16 -> K = 4s+2,4s+3
    // B frag (4x16 f32): lane<16 -> N=lane,  K = 4s+0,4s+1 ; lane>=16 -> K = 4s+2,4s+3
    // Pair-interleaved W => each B frag is ONE b64 load of rows (r0, r0+1).
    v8f accq[HEADS] = {};
    v8f acck[HEADS] = {};
    const int roff = half * 2;
    const float* pAq = &sAq[wave][row * SA_STRIDE + roff];
    const float* pAk = &sAk[wave][row * SA_STRIDE + roff];
    const int colB = lane & 15;
    // pair index pr = r0/2 = 2s + half ; element base = pr*SW2 + 2*col
    const int bbase = half * SW2 + (colB << 1);

#pragma unroll
    for (int s = 0; s < 16; ++s) {
        v2f aQ = *(const v2f*)(pAq + 4 * s);
        v2f aK = *(const v2f*)(pAk + 4 * s);
        const int boff = bbase + s * (2 * SW2);
#pragma unroll
        for (int nt = 0; nt < HEADS; ++nt) {
            v2f bQ = *(const v2f*)&sWQ[boff + nt * 32];
            v2f bK = *(const v2f*)&sWK[boff + nt * 32];
            accq[nt] = __builtin_amdgcn_wmma_f32_16x16x4_f32(
                false, aQ, false, bQ, (short)0, accq[nt], false, false);
            acck[nt] = __builtin_amdgcn_wmma_f32_16x16x4_f32(
                false, aK, false, bK, (short)0, acck[nt], false, false);
        }
    }

    // ---- raw[e,h] = (Q .* K) summed over this head's 16 cols, * 1/sqrt(16) ----
    // C/D layout: VGPR c, lanes0-15 -> M=c ; lanes16-31 -> M=8+c ; N = lane&15.
#pragma unroll
    for (int nt = 0; nt < HEADS; ++nt) {
        v8f p = accq[nt] * acck[nt];
        float r[8];
#pragma unroll
        for (int c = 0; c < 8; ++c) {
            float v = p[c];
            v += __shfl_xor(v, 1);
            v += __shfl_xor(v, 2);
            v += __shfl_xor(v, 4);
            v += __shfl_xor(v, 8);   // masks stay inside each 16-lane half
            r[c] = v * 0.25f;        // 1/sqrt(d_k), d_k = 16
        }
        if ((lane & 15) == 0) {      // lane 0 -> edges base+0..7 ; lane 16 -> base+8..15
#pragma unroll
            for (int c = 0; c < 8; ++c) {
                long e = base + half * 8 + c;
                if (e < E) {
                    raw_out[e * HEADS + nt] = r[c];
                    int it = item_ids[e];
                    atomicMaxF(&seg_max[(long)it * HEADS + nt], r[c]);
                }
            }
        }
    }
}

// ============================================================================
// Workspace init: seg_max = -FLT_MAX, seg_sum = 0, degree = 0 (contiguous).
// ============================================================================
__global__ void kinit(float* __restrict__ ws, long max_region, long total)
{
    long i = (long)blockIdx.x * blockDim.x + threadIdx.x;
    if (i < total) ws[i] = (i < max_region) ? -FLT_MAX : 0.0f;
}

// ============================================================================
// Segment exp-sum + degree count.
// ============================================================================
__global__ void ksum(const float* __restrict__ raw, const int* __restrict__ item_ids,
                     const float* __restrict__ seg_max, float* __restrict__ seg_sum,
                     float* __restrict__ degree, long E)
{
    long i = (long)blockIdx.x * blockDim.x + threadIdx.x;
    if (i >= E * HEADS) return;
    long e = i >> 2;
    int  h = (int)(i & 3);
    int it = item_ids[e];
    float ex = __expf(raw[i] - seg_max[(long)it * HEADS + h]);
    atomicAdd(&seg_sum[(long)it * HEADS + h], ex);
    if (h == 0) atomicAdd(&degree[it], 1.0f);
}

// ============================================================================
// Final: attn = mean_h softmax ; scaled = degree * attn.
// ============================================================================
__global__ void kfinal(const float* __restrict__ raw, const int* __restrict__ item_ids,
                       const float* __restrict__ seg_max, const float* __restrict__ seg_sum,
                       const float* __restrict__ degree,
                       float* __restrict__ attn_out, float* __restrict__ scaled_out, long E)
{
    long e = (long)blockIdx.x * blockDim.x + threadIdx.x;
    if (e >= E) return;
    int it = item_ids[e];
    float s = 0.0f;
#pragma unroll
    for (int h = 0; h < HEADS; ++h)
        s += __expf(raw[e * HEADS + h] - seg_max[(long)it * HEADS + h])
             / seg_sum[(long)it * HEADS + h];
    float a = 0.25f * s;
    attn_out[e]   = a;
    scaled_out[e] = degree[it] * a;
}

extern "C" void kernel_launch(void* const* d_in, const int* in_sizes, int n_in,
                              void* d_out, int out_size, void* d_ws, size_t ws_size,
                              hipStream_t stream)
{
    const int*   item_ids = (const int*)d_in[0];
    const int*   rel_ids  = (const int*)d_in[1];
    const int*   ent_ids  = (const int*)d_in[2];
    const float* ent_embs = (const float*)d_in[3];
    const float* rel_embs = (const float*)d_in[4];
    const float* WQ       = (const float*)d_in[5];
    const float* WK       = (const float*)d_in[6];
    // d_in[7] = n_items (device scalar), d_in[8] = n_heads; D=64, H=4 fixed by layout.

    const long E    = in_sizes[0];
    const long MAXI = in_sizes[3] / D_EMB;   // entity count >= n_items (items are a prefix)

    float* seg_max = (float*)d_ws;                    // [MAXI*H]
    float* seg_sum = seg_max + MAXI * HEADS;          // [MAXI*H]
    float* degree  = seg_sum + MAXI * HEADS;          // [MAXI]

    float* raw_out    = (float*)d_out;                // [E,4]
    float* attn_out   = raw_out + E * HEADS;          // [E]
    float* scaled_out = attn_out + E;                 // [E]

    const long initN = MAXI * (2 * HEADS + 1);
    kinit<<<(int)((initN + 255) / 256), 256, 0, stream>>>(seg_max, MAXI * HEADS, initN);

    const int gemm_blocks = (int)((E + EDGES_PER_BLOCK - 1) / EDGES_PER_BLOCK);
    kgemm_scores<<<gemm_blocks, 128, 0, stream>>>(
        item_ids, rel_ids, ent_ids, ent_embs, rel_embs, WQ, WK,
        raw_out, seg_max, E);

    ksum<<<(int)((E * HEADS + 255) / 256), 256, 0, stream>>>(
        raw_out, item_ids, seg_max, seg_sum, degree, E);

    kfinal<<<(int)((E + 255) / 256), 256, 0, stream>>>(
        raw_out, item_ids, seg_max, seg_sum, degree, attn_out, scaled_out, E);
}